// IntegratedStateSpaceSolver_83708912599087
// MI455X (gfx1250) — compile-verified
//
#include <hip/hip_runtime.h>
#include <hip/hip_bf16.h>
#include <math.h>

#define NBLK 4
#define DLAT 8
#define DIM  16
#define NOBS 4096
#define KST  (2 * NOBS)   /* 8192 */
#define MTEST 8192
#define JIT  1e-6f
#define SQRT3 1.7320508075688772f

typedef __attribute__((ext_vector_type(2))) float v2f;
typedef __attribute__((ext_vector_type(8))) float v8f;

// LDS byte offset of a __shared__ object: LDS aperture occupies a 4GB
// region whose low 32 bits are the LDS offset, so truncation is exact.
#define LDS_OFF(p) ((unsigned)(size_t)(p))

// ------------------------------------------------------------------
// Async global -> LDS copy of one 1KB tile (256 f32) using the CDNA5
// async-DMA path (ASYNCcnt, no VGPR round trip): 32 lanes x b128 x 2.
// ------------------------------------------------------------------
__device__ __forceinline__ void async_copy1k(unsigned lds_dst,
                                             const float* gsrc, int lane)
{
    unsigned voff = (unsigned)lane * 16u;
    unsigned dst  = lds_dst + voff;
    unsigned long long base = (unsigned long long)(size_t)gsrc;
    asm volatile("global_load_async_to_lds_b128 %0, %1, %2 offset:0"
                 :: "v"(dst), "v"(voff), "s"(base) : "memory");
    asm volatile("global_load_async_to_lds_b128 %0, %1, %2 offset:512"
                 :: "v"(dst), "v"(voff), "s"(base) : "memory");
}

__device__ __forceinline__ void async_wait0()
{
    asm volatile("s_wait_asynccnt 0x0" ::: "memory");
}

// ------------------------------------------------------------------
// WMMA helpers: full 16x16 (f32) matmul = chain of 4 x
// v_wmma_f32_16x16x4_f32.  Operand layouts per CDNA5 ISA 7.12.2.
// ------------------------------------------------------------------
template <bool TRANSB>
__device__ __forceinline__ v8f wmma16(const float* A, const float* B,
                                      v8f acc, int lane)
{
    const int r  = lane & 15;
    const int hi = lane >> 4;
#pragma unroll
    for (int kk = 0; kk < 4; ++kk) {
        const int k0 = kk * 4 + 2 * hi;
        v2f a, b;
        a.x = A[r * 16 + k0];
        a.y = A[r * 16 + k0 + 1];
        if (!TRANSB) {
            b.x = B[k0 * 16 + r];
            b.y = B[(k0 + 1) * 16 + r];
        } else {
            b.x = B[r * 16 + k0];
            b.y = B[r * 16 + k0 + 1];
        }
        acc = __builtin_amdgcn_wmma_f32_16x16x4_f32(false, a, false, b,
                                                    (short)0, acc,
                                                    false, false);
    }
    return acc;
}

__device__ __forceinline__ v8f acc_zero() { v8f a = {}; return a; }

__device__ __forceinline__ void acc_store(float* dst, v8f acc, int lane)
{
    const int r = lane & 15, hi = lane >> 4;
#pragma unroll
    for (int v = 0; v < 8; ++v) dst[(v + 8 * hi) * 16 + r] = acc[v];
}

__device__ __forceinline__ v8f acc_load(const float* src, int lane)
{
    const int r = lane & 15, hi = lane >> 4;
    v8f a;
#pragma unroll
    for (int v = 0; v < 8; ++v) a[v] = src[(v + 8 * hi) * 16 + r];
    return a;
}

// ------------------------------------------------------------------
// Closed-form integrated Matern-3/2 transition entries.
// ------------------------------------------------------------------
__device__ __forceinline__ void trans8(float lam, float t,
    float& a11, float& a12, float& a21, float& a22,
    float& i11, float& i12, float& i21, float& i22)
{
    float E  = __expf(-lam * t);
    float lt = lam * t;
    a11 = E * (1.f + lt);
    a12 = t * E;
    a21 = -lam * lam * t * E;
    a22 = E * (1.f - lt);
    float c = 1.f - E * (1.f + lt);
    i11 = (2.f - E * (2.f + lt)) / lam;
    i12 = c / (lam * lam);
    i21 = -c;
    i22 = t * E;
}

__device__ void build_A(float* As, const float* ells, float t, int lane)
{
    if (lane < 16) {
        int r = lane;
#pragma unroll
        for (int j = 0; j < 16; ++j) As[r * 16 + j] = 0.f;
        int b = (r & 7) >> 1;
        float lam = SQRT3 / ells[b];
        float a11, a12, a21, a22, i11, i12, i21, i22;
        trans8(lam, t, a11, a12, a21, a22, i11, i12, i21, i22);
        int cx = 2 * b, cv = 2 * b + 1;
        if (r < 8) {
            if (!(r & 1)) { As[r * 16 + cx] = a11; As[r * 16 + cv] = a12; }
            else          { As[r * 16 + cx] = a21; As[r * 16 + cv] = a22; }
        } else {
            if (!(r & 1)) { As[r * 16 + cx] = i11; As[r * 16 + cv] = i12; }
            else          { As[r * 16 + cx] = i21; As[r * 16 + cv] = i22; }
            As[r * 16 + r] = 1.f;
        }
    }
}

__device__ void build_Q(float* Qs, const float* ells, const float* sigmas,
                        float t, int lane)
{
    if (lane < 16) {
        int r = lane;
#pragma unroll
        for (int j = 0; j < 16; ++j) Qs[r * 16 + j] = 0.f;
        int b = (r & 7) >> 1;
        float lam = SQRT3 / ells[b];
        float sg  = sigmas[b];
        float qc  = 4.f * lam * lam * lam * sg * sg;
        int ri = (r < 8) ? (r & 1) : (2 + (r & 1));
        const float gx[8] = {0.01985507f, 0.10166676f, 0.23723379f, 0.40828268f,
                             0.59171732f, 0.76276621f, 0.89833324f, 0.98014493f};
        const float gw[8] = {0.05061427f, 0.11119052f, 0.15685332f, 0.18134189f,
                             0.18134189f, 0.15685332f, 0.11119052f, 0.05061427f};
        float q0 = 0.f, q1 = 0.f, q2 = 0.f, q3 = 0.f;
#pragma unroll
        for (int q = 0; q < 8; ++q) {
            float u = t * gx[q];
            float a11, a12, a21, a22, i11, i12, i21, i22;
            trans8(lam, u, a11, a12, a21, a22, i11, i12, i21, i22);
            float g0 = a12, g1 = a22, g2 = i12, g3 = i22;
            float gr = (ri == 0) ? g0 : ((ri == 1) ? g1 : ((ri == 2) ? g2 : g3));
            float w = gw[q] * t * qc * gr;
            q0 += w * g0; q1 += w * g1; q2 += w * g2; q3 += w * g3;
        }
        Qs[r * 16 + 2 * b]         += q0;
        Qs[r * 16 + 2 * b + 1]     += q1;
        Qs[r * 16 + 8 + 2 * b]     += q2;
        Qs[r * 16 + 8 + 2 * b + 1] += q3;
        Qs[r * 16 + r] += JIT;
    }
}

// 16x16 Gauss-Jordan inverse (SPD + jitter, no pivoting). Aug = 16x32 LDS.
__device__ void invert16(float* Aug, const float* M, float* Minv, int lane)
{
    for (int i = lane; i < 16 * 32; i += 32) {
        int r = i >> 5, c = i & 31;
        Aug[i] = (c < 16) ? M[r * 16 + c] : ((c - 16) == r ? 1.f : 0.f);
    }
    __syncthreads();
    for (int c = 0; c < 16; ++c) {
        float piv = Aug[c * 32 + c];
        float inv = 1.f / piv;
        __syncthreads();
        Aug[c * 32 + lane] *= inv;
        __syncthreads();
        if (lane < 16 && lane != c) {
            float f = Aug[lane * 32 + c];
            for (int j = 0; j < 32; ++j) Aug[lane * 32 + j] -= f * Aug[c * 32 + j];
        }
        __syncthreads();
    }
    for (int i = lane; i < 256; i += 32) {
        int r = i >> 4, c = i & 15;
        Minv[r * 16 + c] = Aug[r * 32 + 16 + c];
    }
    __syncthreads();
}

// ------------------------------------------------------------------
// Kernel 1: build sort keys (interval endpoints) + payload indices.
// ------------------------------------------------------------------
__global__ void k_init_keys(const float* __restrict__ tmid,
                            const float* __restrict__ delta,
                            float* __restrict__ keys, int* __restrict__ idx)
{
    int i = blockIdx.x * blockDim.x + threadIdx.x;
    if (i < KST) {
        int oid = i >> 1;
        float h = 0.5f * delta[oid];
        keys[i] = (i & 1) ? (tmid[oid] + h) : (tmid[oid] - h);
        idx[i]  = i;   // obsid = idx>>1, stateid = idx&1
    }
}

// Kernel 2: one bitonic compare-exchange step (launched ~91 times).
__global__ void k_bitonic(float* keys, int* idx, int kk, int jj)
{
    int i = blockIdx.x * blockDim.x + threadIdx.x;
    int p = i ^ jj;
    if (i < KST && p > i) {
        float a = keys[i], b = keys[p];
        int ia = idx[i], ib = idx[p];
        bool up = ((i & kk) == 0);
        bool sw = up ? (a > b) : (a < b);
        if (sw) { keys[i] = b; keys[p] = a; idx[i] = ib; idx[p] = ia; }
    }
}

// ------------------------------------------------------------------
// Kernel 3: per-step A(dt), Q(dt).  One wave per step, 8192 blocks.
// ------------------------------------------------------------------
__global__ void __launch_bounds__(32)
k_build_AQ(const float* __restrict__ keys, const float* __restrict__ ells,
           const float* __restrict__ sigmas,
           float* __restrict__ A_all, float* __restrict__ Q_all)
{
    __shared__ float As[256], Qs[256];
    int k = blockIdx.x;
    int lane = threadIdx.x;
    float t = (k == 0) ? 0.f : (keys[k] - keys[k - 1]);
    build_A(As, ells, t, lane);
    build_Q(Qs, ells, sigmas, t, lane);
    __syncthreads();
    for (int i = lane; i < 256; i += 32) {
        A_all[k * 256 + i] = As[i];
        Q_all[k * 256 + i] = Qs[i];
    }
}

// ------------------------------------------------------------------
// Kernel 4: sequential Kalman filter scan.  One wave; A/Q tiles are
// double-buffered in LDS via async global->LDS DMA so the fetch of
// step k+1 overlaps the WMMA chains of step k.
// ------------------------------------------------------------------
__global__ void __launch_bounds__(32)
k_filter(const float* __restrict__ A_all, const float* __restrict__ Q_all,
         const int* __restrict__ idx,
         const float* __restrict__ y, const float* __restrict__ delta,
         const float* __restrict__ sigmas, const float* __restrict__ ells,
         const float* __restrict__ noise_var,
         float* __restrict__ m_pred, float* __restrict__ P_pred,
         float* __restrict__ m_filt, float* __restrict__ P_filt)
{
    __shared__ float Abuf[2][256], Qbuf[2][256];
    __shared__ float Ps[256], Ts[256], Pp[256];
    __shared__ float ms[16], mp[16], PH[16], Kg[16];
    int lane = threadIdx.x;
    const float nv = noise_var[0];

    // init m0, P0
    if (lane < 16) {
        ms[lane] = 0.f;
        for (int j = 0; j < 16; ++j) Ps[lane * 16 + j] = 0.f;
        float v;
        if (lane < 8) {
            int b = lane >> 1;
            float lam = SQRT3 / ells[b];
            float s2 = sigmas[b] * sigmas[b];
            v = (lane & 1) ? lam * lam * s2 : s2;
        } else v = 1e-3f;
        Ps[lane * 16 + lane] = v;
    }
    // prefetch step 0 tiles
    async_copy1k(LDS_OFF(&Abuf[0][0]), A_all, lane);
    async_copy1k(LDS_OFF(&Qbuf[0][0]), Q_all, lane);
    __syncthreads();

    for (int k = 0; k < KST; ++k) {
        const float* As = Abuf[k & 1];
        const float* Qs = Qbuf[k & 1];
        async_wait0();           // step-k tiles landed in LDS
        __syncthreads();
        if (k + 1 < KST) {       // overlap fetch of step k+1 with compute
            async_copy1k(LDS_OFF(&Abuf[(k + 1) & 1][0]),
                         A_all + (size_t)(k + 1) * 256, lane);
            async_copy1k(LDS_OFF(&Qbuf[(k + 1) & 1][0]),
                         Q_all + (size_t)(k + 1) * 256, lane);
        }

        // T = A @ P
        v8f tacc = acc_zero();
        tacc = wmma16<false>(As, Ps, tacc, lane);
        acc_store(Ts, tacc, lane);
        __syncthreads();

        // P_p = T @ A^T + Q   (accumulator preloaded with Q)
        v8f pacc = acc_load(Qs, lane);
        pacc = wmma16<true>(Ts, As, pacc, lane);
        acc_store(Pp, pacc, lane);

        // m_p = A @ m
        if (lane < 16) {
            float s = 0.f;
            for (int j = 0; j < 16; ++j) s += As[lane * 16 + j] * ms[j];
            mp[lane] = s;
        }
        __syncthreads();

        for (int i = lane; i < 256; i += 32) P_pred[k * 256 + i] = Pp[i];
        if (lane < 16) m_pred[k * 16 + lane] = mp[lane];

        int id = idx[k], oid = id >> 1, sid = id & 1;
        if (sid == 0) {
            // interval start: reset integral block
            if (lane < 16) ms[lane] = (lane < 8) ? mp[lane] : 0.f;
            for (int i = lane; i < 256; i += 32) {
                int rr = i >> 4, cc = i & 15;
                Ps[i] = (rr < 8 && cc < 8) ? Pp[i] : 0.f;
            }
        } else {
            // interval end: scalar measurement update with H = H_int/delta
            float dinv = 1.f / delta[oid];
            if (lane < 16) {
                float s = 0.f;
                for (int b = 0; b < 4; ++b) s += Pp[lane * 16 + 8 + 2 * b];
                PH[lane] = s * dinv;
            }
            __syncthreads();
            float S    = (PH[8] + PH[10] + PH[12] + PH[14]) * dinv + nv;
            float vres = y[oid] - (mp[8] + mp[10] + mp[12] + mp[14]) * dinv;
            if (lane < 16) {
                Kg[lane] = PH[lane] / S;
                ms[lane] = mp[lane] + Kg[lane] * vres;
            }
            __syncthreads();
            for (int i = lane; i < 256; i += 32) {
                int rr = i >> 4, cc = i & 15;
                Ps[i] = Pp[i] - Kg[rr] * Kg[cc] * S;
            }
        }
        __syncthreads();
        for (int i = lane; i < 256; i += 32) P_filt[k * 256 + i] = Ps[i];
        if (lane < 16) m_filt[k * 16 + lane] = ms[lane];
        __syncthreads();
    }
}

// ------------------------------------------------------------------
// Kernel 5: sequential RTS smoother (reverse scan).  A_all / P_filt
// tiles double-buffered via async DMA; P_sm/P_pred reads stay
// synchronous (they alias this wave's own prior stores, and async
// loads are unordered against normal stores).
// ------------------------------------------------------------------
__global__ void __launch_bounds__(32)
k_smooth(const float* __restrict__ A_all,
         const float* __restrict__ m_pred, const float* __restrict__ P_pred,
         const float* __restrict__ m_filt, const float* __restrict__ P_filt,
         float* __restrict__ m_sm, float* __restrict__ P_sm)
{
    __shared__ float Abuf[2][256], Pfbuf[2][256];
    __shared__ float Ts[256], Gs[256], Dp[256], Pinv[256];
    __shared__ float Aug[512];
    __shared__ float dm[16];
    int lane = threadIdx.x;

    for (int i = lane; i < 256; i += 32)
        P_sm[(size_t)(KST - 1) * 256 + i] = P_filt[(size_t)(KST - 1) * 256 + i];
    if (lane < 16) m_sm[(KST - 1) * 16 + lane] = m_filt[(KST - 1) * 16 + lane];

    // prefetch tiles for k = KST-2
    async_copy1k(LDS_OFF(&Abuf[(KST - 2) & 1][0]),
                 A_all + (size_t)(KST - 1) * 256, lane);
    async_copy1k(LDS_OFF(&Pfbuf[(KST - 2) & 1][0]),
                 P_filt + (size_t)(KST - 2) * 256, lane);
    __syncthreads();

    for (int k = KST - 2; k >= 0; --k) {
        const float* As = Abuf[k & 1];
        const float* Pf = Pfbuf[k & 1];
        async_wait0();
        __syncthreads();
        if (k - 1 >= 0) {   // overlap next iteration's tiles with compute
            async_copy1k(LDS_OFF(&Abuf[(k - 1) & 1][0]),
                         A_all + (size_t)k * 256, lane);
            async_copy1k(LDS_OFF(&Pfbuf[(k - 1) & 1][0]),
                         P_filt + (size_t)(k - 1) * 256, lane);
        }

        invert16(Aug, P_pred + (size_t)(k + 1) * 256, Pinv, lane);

        // T = P_f @ A^T
        v8f t1 = acc_zero();
        t1 = wmma16<true>(Pf, As, t1, lane);
        acc_store(Ts, t1, lane);
        __syncthreads();
        // G = T @ Pinv
        v8f g = acc_zero();
        g = wmma16<false>(Ts, Pinv, g, lane);
        acc_store(Gs, g, lane);
        if (lane < 16)
            dm[lane] = m_sm[(k + 1) * 16 + lane] - m_pred[(k + 1) * 16 + lane];
        __syncthreads();

        if (lane < 16) {
            float s = 0.f;
            for (int j = 0; j < 16; ++j) s += Gs[lane * 16 + j] * dm[j];
            m_sm[k * 16 + lane] = m_filt[k * 16 + lane] + s;
        }
        for (int i = lane; i < 256; i += 32)
            Dp[i] = P_sm[(size_t)(k + 1) * 256 + i] - P_pred[(size_t)(k + 1) * 256 + i];
        __syncthreads();

        // P_s = P_f + G @ Dp @ G^T
        v8f t2 = acc_zero();
        t2 = wmma16<false>(Gs, Dp, t2, lane);
        acc_store(Ts, t2, lane);
        __syncthreads();
        v8f t3 = acc_zero();
        t3 = wmma16<true>(Ts, Gs, t3, lane);
        {
            const int r = lane & 15, hi = lane >> 4;
#pragma unroll
            for (int v = 0; v < 8; ++v) {
                int e = (v + 8 * hi) * 16 + r;
                P_sm[(size_t)k * 256 + e] = Pf[e] + t3[v];
            }
        }
        __syncthreads();
    }
}

// ------------------------------------------------------------------
// Kernel 6: predictions. One wave per test point; wave-uniform case
// selection keeps EXEC full around the WMMA chains.
// ------------------------------------------------------------------
__global__ void __launch_bounds__(32)
k_predict(const float* __restrict__ keys, const float* __restrict__ t_test,
          const float* __restrict__ ells, const float* __restrict__ sigmas,
          const float* __restrict__ m_pred, const float* __restrict__ P_pred,
          const float* __restrict__ m_filt, const float* __restrict__ P_filt,
          const float* __restrict__ m_sm,   const float* __restrict__ P_sm,
          float* __restrict__ out_mean, float* __restrict__ out_var)
{
    __shared__ float As[256], Qs[256], Psp[256], Ts[256], Gs[256], Dp[256], Pinv[256];
    __shared__ float Aug[512];
    __shared__ float msp[16], dm[16], mres[16];
    int lane = threadIdx.x;
    int j = blockIdx.x;
    float tt = t_test[j];

    // searchsorted(keys, tt, side='right')
    int lo = 0, hiK = KST;
    while (lo < hiK) {
        int mid = (lo + hiK) >> 1;
        if (keys[mid] <= tt) lo = mid + 1; else hiK = mid;
    }
    int kn = lo;
    bool s1 = (kn > 0);      // Kalman propagate stage
    bool s2 = (kn < KST);    // smoothing stage

    // warm L2/L1 for the tiles this block will consume
    if (s2) {
        __builtin_prefetch((const void*)(P_pred + (size_t)kn * 256), 0, 3);
        __builtin_prefetch((const void*)(P_sm + (size_t)kn * 256), 0, 3);
    }
    if (s1) {
        __builtin_prefetch((const void*)(P_filt + (size_t)(kn - 1) * 256), 0, 3);
    }

    if (s1) {
        int kb = kn - 1;
        float dt = tt - keys[kb];
        build_A(As, ells, dt, lane);
        build_Q(Qs, ells, sigmas, dt, lane);
        for (int i = lane; i < 256; i += 32) Ts[i] = P_filt[(size_t)kb * 256 + i];
        __syncthreads();
        // Psp = A @ Pf @ A^T + Q
        v8f t1 = acc_zero();
        t1 = wmma16<false>(As, Ts, t1, lane);
        acc_store(Dp, t1, lane);
        __syncthreads();
        v8f p = acc_load(Qs, lane);
        p = wmma16<true>(Dp, As, p, lane);
        acc_store(Psp, p, lane);
        if (lane < 16) {
            float s = 0.f;
            for (int jj = 0; jj < 16; ++jj)
                s += As[lane * 16 + jj] * m_filt[kb * 16 + jj];
            msp[lane] = s;
        }
        __syncthreads();
    } else {
        // prior (m0, P0)
        if (lane < 16) {
            msp[lane] = 0.f;
            for (int jj = 0; jj < 16; ++jj) Psp[lane * 16 + jj] = 0.f;
            float v;
            if (lane < 8) {
                int b = lane >> 1;
                float lam = SQRT3 / ells[b];
                float s2 = sigmas[b] * sigmas[b];
                v = (lane & 1) ? lam * lam * s2 : s2;
            } else v = 1e-3f;
            Psp[lane * 16 + lane] = v;
        }
        __syncthreads();
    }

    if (s2) {
        int ks = kn;
        float dt2 = keys[ks] - tt;
        build_A(As, ells, dt2, lane);
        __syncthreads();
        invert16(Aug, P_pred + (size_t)ks * 256, Pinv, lane);

        // G = Psp @ A^T @ inv(P_pred[ks])
        v8f t1 = acc_zero();
        t1 = wmma16<true>(Psp, As, t1, lane);
        acc_store(Ts, t1, lane);
        __syncthreads();
        v8f g = acc_zero();
        g = wmma16<false>(Ts, Pinv, g, lane);
        acc_store(Gs, g, lane);
        if (lane < 16)
            dm[lane] = m_sm[ks * 16 + lane] - m_pred[ks * 16 + lane];
        __syncthreads();

        if (lane < 16) {
            float s = 0.f;
            for (int jj = 0; jj < 16; ++jj) s += Gs[lane * 16 + jj] * dm[jj];
            mres[lane] = msp[lane] + s;
        }
        for (int i = lane; i < 256; i += 32)
            Dp[i] = P_sm[(size_t)ks * 256 + i] - P_pred[(size_t)ks * 256 + i];
        __syncthreads();

        // P = Psp + G @ Dp @ G^T
        v8f t2 = acc_zero();
        t2 = wmma16<false>(Gs, Dp, t2, lane);
        acc_store(Ts, t2, lane);
        __syncthreads();
        v8f t3 = acc_zero();
        t3 = wmma16<true>(Ts, Gs, t3, lane);
        {
            const int r = lane & 15, hi = lane >> 4;
#pragma unroll
            for (int v = 0; v < 8; ++v) {
                int e = (v + 8 * hi) * 16 + r;
                Psp[e] = Psp[e] + t3[v];
            }
        }
        __syncthreads();
        if (lane < 16) msp[lane] = mres[lane];
        __syncthreads();
    }

    // project: H_latent picks x-components (even indices of first 8)
    if (lane == 0) {
        float mean = msp[0] + msp[2] + msp[4] + msp[6];
        float var = 0.f;
        for (int a = 0; a < 8; a += 2)
            for (int b = 0; b < 8; b += 2)
                var += Psp[a * 16 + b];
        out_mean[j] = mean;
        out_var[j]  = var;
    }
}

// ------------------------------------------------------------------
extern "C" void kernel_launch(void* const* d_in, const int* in_sizes, int n_in,
                              void* d_out, int out_size, void* d_ws, size_t ws_size,
                              hipStream_t stream)
{
    (void)in_sizes; (void)n_in; (void)out_size; (void)ws_size;
    const float* tmid   = (const float*)d_in[0];
    const float* delta  = (const float*)d_in[1];
    const float* y      = (const float*)d_in[2];
    const float* t_test = (const float*)d_in[3];
    const float* sigmas = (const float*)d_in[4];
    const float* ells   = (const float*)d_in[5];
    const float* noise  = (const float*)d_in[6];
    float* out = (float*)d_out;

    char* ws = (char*)d_ws;
    float* keys   = (float*)(ws);                     // KST
    int*   idx    = (int*)(ws + KST * 4);             // KST
    float* A_all  = (float*)(ws + 2 * KST * 4);       // KST*256
    float* Q_all  = A_all + (size_t)KST * 256;
    float* m_pred = Q_all + (size_t)KST * 256;
    float* P_pred = m_pred + (size_t)KST * 16;
    float* m_filt = P_pred + (size_t)KST * 256;
    float* P_filt = m_filt + (size_t)KST * 16;
    float* m_sm   = P_filt + (size_t)KST * 256;
    float* P_sm   = m_sm + (size_t)KST * 16;

    // 1) keys + payload
    k_init_keys<<<KST / 256, 256, 0, stream>>>(tmid, delta, keys, idx);

    // 2) bitonic sort of 8192 endpoint times (91 compare-exchange passes)
    for (int kk = 2; kk <= KST; kk <<= 1)
        for (int jj = kk >> 1; jj > 0; jj >>= 1)
            k_bitonic<<<KST / 256, 256, 0, stream>>>(keys, idx, kk, jj);

    // 3) per-step transition + process noise (parallel, closed form)
    k_build_AQ<<<KST, 32, 0, stream>>>(keys, ells, sigmas, A_all, Q_all);

    // 4) forward Kalman scan (sequential, WMMA 16x16, async double-buffer)
    k_filter<<<1, 32, 0, stream>>>(A_all, Q_all, idx, y, delta, sigmas, ells,
                                   noise, m_pred, P_pred, m_filt, P_filt);

    // 5) backward RTS scan (sequential, WMMA 16x16 + GJ inverse, async DB)
    k_smooth<<<1, 32, 0, stream>>>(A_all, m_pred, P_pred, m_filt, P_filt,
                                   m_sm, P_sm);

    // 6) 8192 independent predictions (one wave each, WMMA 16x16)
    k_predict<<<MTEST, 32, 0, stream>>>(keys, t_test, ells, sigmas,
                                        m_pred, P_pred, m_filt, P_filt,
                                        m_sm, P_sm, out, out + MTEST);
}